// FAGCN_75496935129291
// MI455X (gfx1250) — compile-verified
//
#include <hip/hip_runtime.h>
#include <hip/hip_bf16.h>

// ---------------------------------------------------------------------------
// FAGCN forward on MI455X (gfx1250), wave32, WMMA f32 16x16x4.
//   x    = relu(h @ t1_w^T + t1_b)                       (WMMA GEMM)
//   per layer: aD[n]=<x[n],gw[0:64]>, aS[n]=<x[n],gw[64:128]>   (wave reduce)
//              e_ij = tanh(aD[dst]+aS[src]+gb) * d[dst]*d[src]  (edge kernel)
//              z[dst] += x[src]*e                         (hw f32 atomics)
//              x = EPS*raw + z
//   out  = log_softmax(x @ t2_w^T + t2_b)                 (WMMA + shfl softmax)
// ---------------------------------------------------------------------------

typedef float v2f __attribute__((ext_vector_type(2)));
typedef float v8f __attribute__((ext_vector_type(8)));

#define IN_F   256
#define HID    64
#define OUT_F  16
#define EPS_C  0.3f

// hardware global_atomic_add_f32 (no CAS loop); order-insensitive segment sum
__device__ __forceinline__ void atomic_fadd_hw(float* p, float v) {
#if defined(__HIP_DEVICE_COMPILE__)
  unsafeAtomicAdd(p, v);
#else
  atomicAdd(p, v);
#endif
}

// ---------------- utility ----------------
__global__ void zero_f32(float* __restrict__ p, int n) {
  int i = blockIdx.x * blockDim.x + threadIdx.x;
  if (i < n) p[i] = 0.0f;
}

__global__ void deg_scatter(const int* __restrict__ dst, float* __restrict__ deg, int E) {
  int i = blockIdx.x * blockDim.x + threadIdx.x;
  if (i < E) atomic_fadd_hw(&deg[dst[i]], 1.0f);
}

__global__ void deg_rsqrt(const float* __restrict__ deg, float* __restrict__ d, int N) {
  int i = blockIdx.x * blockDim.x + threadIdx.x;
  if (i < N) d[i] = rsqrtf(fmaxf(deg[i], 1.0f));
}

// ---------------- GEMM1: x = relu(h @ t1_w^T + b), also writes raw ----------
// One wave per 16x16 C tile. K = 256 stepped by 4 (v_wmma_f32_16x16x4_f32).
// A (16x4 f32): lanes 0-15 hold M=lane {K=k,k+1}; lanes 16-31 hold {K=k+2,k+3}
// B (4x16 f32): lanes 0-15 hold N=lane rows {K=k,k+1}; lanes 16-31 {k+2,k+3}
// C (16x16 f32): VGPR r -> M=r (lanes 0-15) / M=r+8 (lanes 16-31), N=lane&15
__global__ void gemm1_relu(const float* __restrict__ h,
                           const float* __restrict__ t1_w,   // [64,256]
                           const float* __restrict__ t1_b,   // [64]
                           float* __restrict__ x,
                           float* __restrict__ raw,
                           int N) {
  const int wave = (blockIdx.x * blockDim.x + threadIdx.x) >> 5;
  const int lane = threadIdx.x & 31;
  const int tilesN = HID / 16;                 // 4
  const int totalTiles = (N / 16) * tilesN;
  if (wave >= totalTiles) return;

  const int m0 = (wave / tilesN) * 16;
  const int n0 = (wave % tilesN) * 16;

  const int half   = (lane < 16) ? 0 : 2;      // K sub-offset for this half
  const int mRow   = m0 + (lane & 15);         // A row this lane feeds
  const int nCol   = n0 + (lane & 15);         // B col this lane feeds

  v8f c = {};
  for (int k = 0; k < IN_F; k += 4) {
    v2f a = *reinterpret_cast<const v2f*>(&h[mRow * IN_F + k + half]);
    v2f b = *reinterpret_cast<const v2f*>(&t1_w[nCol * IN_F + k + half]);
    c = __builtin_amdgcn_wmma_f32_16x16x4_f32(false, a, false, b,
                                              (short)0, c, false, false);
  }

  const float bias = t1_b[nCol];
  const int   mBase = m0 + ((lane < 16) ? 0 : 8);
#pragma unroll
  for (int r = 0; r < 8; ++r) {
    float v = fmaxf(c[r] + bias, 0.0f);
    int row = mBase + r;
    x[row * HID + nCol]   = v;
    raw[row * HID + nCol] = v;
  }
}

// ---------------- per-node gate dot products --------------------------------
// one wave per node: aD[n] = <x[n], gw[0:64]>, aS[n] = <x[n], gw[64:128]>
__global__ void gate_dots(const float* __restrict__ x,
                          const float* __restrict__ gw,   // [128] = gate_w[l]
                          float* __restrict__ aD,
                          float* __restrict__ aS,
                          int N) {
  const int wave = (blockIdx.x * blockDim.x + threadIdx.x) >> 5;
  const int lane = threadIdx.x & 31;
  if (wave >= N) return;

  v2f xv = *reinterpret_cast<const v2f*>(&x[wave * HID + lane * 2]);
  v2f wd = *reinterpret_cast<const v2f*>(&gw[lane * 2]);
  v2f ws = *reinterpret_cast<const v2f*>(&gw[HID + lane * 2]);
  float sd = xv.x * wd.x + xv.y * wd.y;
  float ss = xv.x * ws.x + xv.y * ws.y;
#pragma unroll
  for (int off = 16; off > 0; off >>= 1) {
    sd += __shfl_xor(sd, off, 32);
    ss += __shfl_xor(ss, off, 32);
  }
  if (lane == 0) { aD[wave] = sd; aS[wave] = ss; }
}

// ---------------- per-edge coefficient --------------------------------------
__global__ void edge_coef(const int* __restrict__ src, const int* __restrict__ dst,
                          const float* __restrict__ aD, const float* __restrict__ aS,
                          const float* __restrict__ d,
                          const float* __restrict__ gb, int l,
                          float* __restrict__ e, int E) {
  int i = blockIdx.x * blockDim.x + threadIdx.x;
  if (i >= E) return;
  int s = src[i], t = dst[i];
  float g = tanhf(aD[t] + aS[s] + gb[l]);
  e[i] = g * d[t] * d[s];
}

// ---------------- message scatter: z[dst] += x[src] * e ---------------------
// thread = (edge, feat); 64 consecutive threads share one edge -> coalesced
__global__ void msg_scatter(const int* __restrict__ src, const int* __restrict__ dst,
                            const float* __restrict__ x, const float* __restrict__ e,
                            float* __restrict__ z, int E) {
  long long idx = (long long)blockIdx.x * blockDim.x + threadIdx.x;
  int edge = (int)(idx >> 6);
  int feat = (int)(idx & 63);
  if (edge >= E) return;
  float ev = e[edge];
  float v  = x[(long long)src[edge] * HID + feat] * ev;
  atomic_fadd_hw(&z[(long long)dst[edge] * HID + feat], v);
}

// ---------------- residual combine: x = EPS*raw + z -------------------------
__global__ void combine(const float* __restrict__ raw, const float* __restrict__ z,
                        float* __restrict__ x, int n) {
  int i = blockIdx.x * blockDim.x + threadIdx.x;
  if (i < n) x[i] = EPS_C * raw[i] + z[i];
}

// ---------------- GEMM2 + log_softmax ---------------------------------------
// One wave per 16-row tile, single 16-col tile (OUT=16), K=64 by 4.
// Row m lives across a 16-lane half in VGPR r -> shfl_xor(width 16) softmax.
__global__ void gemm2_logsoftmax(const float* __restrict__ x,
                                 const float* __restrict__ t2_w,  // [16,64]
                                 const float* __restrict__ t2_b,  // [16]
                                 float* __restrict__ out,
                                 int N) {
  const int wave = (blockIdx.x * blockDim.x + threadIdx.x) >> 5;
  const int lane = threadIdx.x & 31;
  const int totalTiles = N / 16;
  if (wave >= totalTiles) return;

  const int m0   = wave * 16;
  const int half = (lane < 16) ? 0 : 2;
  const int mRow = m0 + (lane & 15);
  const int nCol = lane & 15;

  v8f c = {};
  for (int k = 0; k < HID; k += 4) {
    v2f a = *reinterpret_cast<const v2f*>(&x[mRow * HID + k + half]);
    v2f b = *reinterpret_cast<const v2f*>(&t2_w[nCol * HID + k + half]);
    c = __builtin_amdgcn_wmma_f32_16x16x4_f32(false, a, false, b,
                                              (short)0, c, false, false);
  }

  const float bias  = t2_b[nCol];
  const int   mBase = m0 + ((lane < 16) ? 0 : 8);
#pragma unroll
  for (int r = 0; r < 8; ++r) {
    float v = c[r] + bias;
    float mx = v;
#pragma unroll
    for (int off = 8; off > 0; off >>= 1) mx = fmaxf(mx, __shfl_xor(mx, off, 16));
    float s = __expf(v - mx);
#pragma unroll
    for (int off = 8; off > 0; off >>= 1) s += __shfl_xor(s, off, 16);
    out[(mBase + r) * OUT_F + nCol] = v - mx - __logf(s);
  }
}

// ---------------------------------------------------------------------------
extern "C" void kernel_launch(void* const* d_in, const int* in_sizes, int n_in,
                              void* d_out, int out_size, void* d_ws, size_t ws_size,
                              hipStream_t stream) {
  const float* h      = (const float*)d_in[0];
  const int*   src    = (const int*)  d_in[1];
  const int*   dst    = (const int*)  d_in[2];
  const float* t1_w   = (const float*)d_in[3];
  const float* t1_b   = (const float*)d_in[4];
  const float* gate_w = (const float*)d_in[5];   // [2, 128]
  const float* gate_b = (const float*)d_in[6];   // [2]
  const float* t2_w   = (const float*)d_in[7];
  const float* t2_b   = (const float*)d_in[8];
  float* out = (float*)d_out;

  const int N  = in_sizes[0] / IN_F;             // 50000
  const int E  = in_sizes[1];                    // 800000
  const int NH = N * HID;

  float* ws  = (float*)d_ws;
  float* raw = ws;                // N*H
  float* x   = raw + NH;          // N*H
  float* z   = x   + NH;          // N*H  (also reused as deg temp)
  float* d   = z   + NH;          // N
  float* aD  = d   + N;           // N
  float* aS  = aD  + N;           // N
  float* ec  = aS  + N;           // E

  const int B = 256;

  // degree -> d = rsqrt(max(deg,1))
  zero_f32<<<(N + B - 1) / B, B, 0, stream>>>(z, N);
  deg_scatter<<<(E + B - 1) / B, B, 0, stream>>>(dst, z, E);
  deg_rsqrt<<<(N + B - 1) / B, B, 0, stream>>>(z, d, N);

  // GEMM1 + relu (writes x and raw)
  {
    int tiles = (N / 16) * (HID / 16);
    int waves = tiles, blocks = (waves * 32 + B - 1) / B;
    gemm1_relu<<<blocks, B, 0, stream>>>(h, t1_w, t1_b, x, raw, N);
  }

  for (int l = 0; l < 2; ++l) {
    gate_dots<<<(N * 32 + B - 1) / B, B, 0, stream>>>(x, gate_w + l * 2 * HID, aD, aS, N);
    edge_coef<<<(E + B - 1) / B, B, 0, stream>>>(src, dst, aD, aS, d, gate_b, l, ec, E);
    zero_f32<<<(NH + B - 1) / B, B, 0, stream>>>(z, NH);
    {
      long long tot = (long long)E * HID;
      int blocks = (int)((tot + B - 1) / B);
      msg_scatter<<<blocks, B, 0, stream>>>(src, dst, x, ec, z, E);
    }
    combine<<<(NH + B - 1) / B, B, 0, stream>>>(raw, z, x, NH);
  }

  // GEMM2 + log_softmax -> out
  {
    int waves = N / 16, blocks = (waves * 32 + B - 1) / B;
    gemm2_logsoftmax<<<blocks, B, 0, stream>>>(x, t2_w, t2_b, out, N);
  }
}